// GraphAttentionLayer_25099788878408
// MI455X (gfx1250) — compile-verified
//
#include <hip/hip_runtime.h>
#include <hip/hip_bf16.h>

// GAT layer: B=4, D=4096, F_IN=F_OUT=128, leaky slope 0.2
// Memory-bound (adj read + attn write = 536 MB floor). Strategy:
//  - full f32 precision, V_WMMA_F32_16X16X4_F32 for both GEMMs
//  - FUSE softmax + attn@Wh: 16 full softmax rows (256 KB) live in the
//    320 KB WGP LDS, killing the 268 MB attn re-read of a 2-kernel version
//  - softmax normalization folded into the GEMM epilogue (scale by 1/sum)
//  - Wh kept TRANSPOSED (WhT[b][n][k]) so the WMMA B-operand is one
//    contiguous b64 load per wmma instead of two strided b32 loads

#define GAT_B 4
#define GAT_D 4096
#define GAT_F 128
#define GAT_ROWS (GAT_B * GAT_D)   // 16384
#define SLOPE 0.2f
#define LDS_STRIDE (GAT_D + 4)     // 4100: 4-bank skew per row -> conflict-free

typedef float v2f __attribute__((ext_vector_type(2)));
typedef float v8f __attribute__((ext_vector_type(8)));

#if defined(__gfx1250__) && __has_builtin(__builtin_amdgcn_wmma_f32_16x16x4_f32)
#define HAVE_WMMA_F32X4 1
#endif

// D(16x16,f32) = A(16x4,f32) * B(4x16,f32) + C   (wave32 lane layouts)
//   A: lane l holds A[l%16][ (l/16)*2 + v ] in component v (contiguous K pair)
//   B: lane l holds B[ (l/16)*2 + v ][ l%16 ] in component v
//   C/D: component r is element [ r + (l/16)*8 ][ l%16 ]
__device__ __forceinline__ v8f wmma_f32_16x16x4(v2f a, v2f b, v8f c) {
#ifdef HAVE_WMMA_F32X4
    return __builtin_amdgcn_wmma_f32_16x16x4_f32(
        /*neg_a=*/false, a, /*neg_b=*/false, b,
        /*c_mod=*/(short)0, c, /*reuse_a=*/false, /*reuse_b=*/false);
#else
    c[0] += a.x * b.x + a.y * b.y;   // compile-only fallback
    return c;
#endif
}

// ---------------------------------------------------------------------------
// K1: WhT[b][n][i] = sum_k x[b,i,k] * W[k,n];  one wave per 16x16 tile.
// Stored transposed so downstream B-operand reads are K-contiguous.
// grid = (ROWS/16, F/16), block = 32
// ---------------------------------------------------------------------------
__global__ __launch_bounds__(32) void wh_gemm_kernel(
    const float* __restrict__ x, const float* __restrict__ W,
    float* __restrict__ WhT) {
    const int lane = threadIdx.x & 31;
    const int half = lane >> 4;
    const int mrow = lane & 15;
    const long row0  = (long)blockIdx.x * 16;      // global row (b*D + i)
    const long b     = row0 >> 12;
    const long ibase = row0 & (GAT_D - 1);
    const int  n0    = blockIdx.y * 16;

    const float* arow = x + (row0 + mrow) * GAT_F;
    v8f acc = {};
#pragma unroll
    for (int k0 = 0; k0 < GAT_F; k0 += 4) {
        const int ka = k0 + half * 2;
        v2f a = *(const v2f*)(arow + ka);
        v2f bb;
        bb.x = W[(ka + 0) * GAT_F + n0 + mrow];
        bb.y = W[(ka + 1) * GAT_F + n0 + mrow];
        acc = wmma_f32_16x16x4(a, bb, acc);
    }
    // transposed store: WhT[(b*F + n) * D + i]
    float* orow = WhT + (b * GAT_F + n0 + mrow) * (long)GAT_D + ibase;
#pragma unroll
    for (int r = 0; r < 8; ++r) {
        orow[r + half * 8] = acc[r];
    }
}

// ---------------------------------------------------------------------------
// K2: f_src[row] = dot(Wh[row,:], a_src); f_tgt likewise. One thread per row.
// Reads WhT: consecutive threads hit consecutive addresses at each k.
// ---------------------------------------------------------------------------
__global__ __launch_bounds__(256) void fproj_kernel(
    const float* __restrict__ WhT, const float* __restrict__ a_src,
    const float* __restrict__ a_tgt, float* __restrict__ f_src,
    float* __restrict__ f_tgt) {
    const long row = (long)blockIdx.x * blockDim.x + threadIdx.x;
    if (row >= GAT_ROWS) return;
    const long b = row >> 12;
    const long i = row & (GAT_D - 1);
    const float* w = WhT + b * GAT_F * (long)GAT_D + i;
    float s = 0.f, t = 0.f;
#pragma unroll 8
    for (int k = 0; k < GAT_F; ++k) {
        const float v = w[(long)k * GAT_D];
        s += v * a_src[k];
        t += v * a_tgt[k];
    }
    f_src[row] = s;
    f_tgt[row] = t;
}

// ---------------------------------------------------------------------------
// K3 (fused): per-row softmax of masked leaky scores + h' = attn@Wh + bias.
// One block = 16 rows. 256 threads = 8 waves.
//  Phase 1: wave w softmaxes rows 2w, 2w+1 entirely in LDS (wave-local
//           shuffle reductions only), writes normalized attn to global,
//           keeps UNNORMALIZED exp values in LDS, stashes 1/sum in inv16[].
//  Phase 2: 8 waves x one 16x16 N-tile each; A = exp values from LDS (b64),
//           B = WhT streamed from L2 (one b64 per wmma);
//           epilogue scales by inv16[m] and adds bias.
// Dynamic LDS: 16 * 4100 * 4 B = 256.6 KB (fits CDNA5's 320 KB WGP LDS).
// ---------------------------------------------------------------------------
__global__ __launch_bounds__(256) void fused_softmax_gemm_kernel(
    const int* __restrict__ adj, const float* __restrict__ f_src,
    const float* __restrict__ f_tgt, const float* __restrict__ WhT,
    const float* __restrict__ bias, float* __restrict__ attn,
    float* __restrict__ hp) {
    extern __shared__ float ex_s[];          // [16][LDS_STRIDE]
    __shared__ float inv16[16];

    const int tid  = threadIdx.x;
    const int wave = tid >> 5;
    const int lane = tid & 31;
    const long grow0 = (long)blockIdx.x * 16;   // first global row (b*D + i)
    const long b     = grow0 >> 12;
    const long bbase = b << 12;                 // b*D
    const float* ft = f_tgt + bbase;

    // -------- phase 1: softmax for this wave's two rows --------
#pragma unroll
    for (int rr = 0; rr < 2; ++rr) {
        const int  rloc = wave * 2 + rr;
        const long grow = grow0 + rloc;
        const float fs  = f_src[grow];
        const int* arow = adj + grow * (long)GAT_D;
        float* row_s    = ex_s + rloc * LDS_STRIDE;

        if (rr == 0)  // warm next row's adj stream (global_prefetch_b8)
            __builtin_prefetch(adj + (grow + 1) * (long)GAT_D + lane * 4, 0, 0);

        // masked leaky-relu scores -> LDS, track max
        float lmax = -3.402823e38f;
        for (int j = lane * 4; j < GAT_D; j += 128) {
            const int4   m = *(const int4*)(arow + j);
            const float4 f = *(const float4*)(ft + j);
            float4 e;
            e.x = fs + f.x; e.x = (e.x > 0.f) ? e.x : SLOPE * e.x; e.x = m.x ? e.x : 0.f;
            e.y = fs + f.y; e.y = (e.y > 0.f) ? e.y : SLOPE * e.y; e.y = m.y ? e.y : 0.f;
            e.z = fs + f.z; e.z = (e.z > 0.f) ? e.z : SLOPE * e.z; e.z = m.z ? e.z : 0.f;
            e.w = fs + f.w; e.w = (e.w > 0.f) ? e.w : SLOPE * e.w; e.w = m.w ? e.w : 0.f;
            *(float4*)(row_s + j) = e;
            lmax = fmaxf(lmax, fmaxf(fmaxf(e.x, e.y), fmaxf(e.z, e.w)));
        }
#pragma unroll
        for (int off = 16; off > 0; off >>= 1)
            lmax = fmaxf(lmax, __shfl_xor(lmax, off, 32));

        // exponentiate in place, accumulate sum
        float lsum = 0.f;
        for (int j = lane * 4; j < GAT_D; j += 128) {
            float4 e = *(float4*)(row_s + j);
            e.x = __expf(e.x - lmax);
            e.y = __expf(e.y - lmax);
            e.z = __expf(e.z - lmax);
            e.w = __expf(e.w - lmax);
            *(float4*)(row_s + j) = e;
            lsum += (e.x + e.y) + (e.z + e.w);
        }
#pragma unroll
        for (int off = 16; off > 0; off >>= 1)
            lsum += __shfl_xor(lsum, off, 32);
        const float inv = 1.f / lsum;
        if (lane == 0) inv16[rloc] = inv;

        // write normalized attention row (softmax output is already in [0,1])
        float* orow = attn + grow * (long)GAT_D;
        for (int j = lane * 4; j < GAT_D; j += 128) {
            float4 e = *(const float4*)(row_s + j);
            e.x *= inv; e.y *= inv; e.z *= inv; e.w *= inv;
            *(float4*)(orow + j) = e;
        }
    }
    __syncthreads();

    // -------- phase 2: 16x128 GEMM on unnormalized exp values --------
    const int half = lane >> 4;
    const int mrow = lane & 15;
    const int n0   = wave * 16;
    const float* rowA = ex_s + mrow * LDS_STRIDE;
    const float* rowB = WhT + (b * GAT_F + n0 + mrow) * (long)GAT_D;  // K-contig

    v8f acc = {};
#pragma unroll 8
    for (int k0 = 0; k0 < GAT_D; k0 += 4) {
        const int ka = k0 + half * 2;
        v2f a = *(const v2f*)(rowA + ka);   // ds_load_b64
        v2f bb = *(const v2f*)(rowB + ka);  // global_load_b64 (K-contiguous)
        acc = wmma_f32_16x16x4(a, bb, acc);
    }

    const float bz = bias[n0 + mrow];
#pragma unroll
    for (int r = 0; r < 8; ++r) {
        const int m = r + half * 8;
        hp[(grow0 + m) * GAT_F + n0 + mrow] = acc[r] * inv16[m] + bz;
    }
}

// ---------------------------------------------------------------------------
extern "C" void kernel_launch(void* const* d_in, const int* in_sizes, int n_in,
                              void* d_out, int out_size, void* d_ws, size_t ws_size,
                              hipStream_t stream) {
    const float* x     = (const float*)d_in[0];  // [B,D,F]
    const int*   adj   = (const int*)  d_in[1];  // [B,D,D]
    const float* W     = (const float*)d_in[2];  // [F,F]
    const float* a_src = (const float*)d_in[3];  // [F]
    const float* a_tgt = (const float*)d_in[4];  // [F]
    const float* bias  = (const float*)d_in[5];  // [F]

    float* out     = (float*)d_out;
    float* h_prime = out;                                  // B*D*F floats
    float* attn    = out + (size_t)GAT_ROWS * GAT_F;       // B*D*D floats

    float* WhT   = (float*)d_ws;                           // B*F*D floats (8 MB)
    float* f_src = WhT + (size_t)GAT_B * GAT_F * GAT_D;    // B*D floats
    float* f_tgt = f_src + (size_t)GAT_ROWS;               // B*D floats

    // K1: WhT = (x @ W)^T per batch
    wh_gemm_kernel<<<dim3(GAT_ROWS / 16, GAT_F / 16), 32, 0, stream>>>(x, W, WhT);
    // K2: per-row attention projections
    fproj_kernel<<<GAT_ROWS / 256, 256, 0, stream>>>(WhT, a_src, a_tgt, f_src, f_tgt);
    // K3: fused masked-softmax + attn@Wh + bias (writes both outputs)
    const size_t lds_bytes = (size_t)16 * LDS_STRIDE * sizeof(float);  // ~256.6 KB
    fused_softmax_gemm_kernel<<<GAT_ROWS / 16, 256, lds_bytes, stream>>>(
        adj, f_src, f_tgt, WhT, bias, attn, h_prime);
}